// SSM_2104533975587
// MI455X (gfx1250) — compile-verified
//
#include <hip/hip_runtime.h>
#include <stdint.h>

// ---------------------------------------------------------------- constants
#define B_   2
#define L_   2048
#define DM_  1024
#define DI_  2048
#define DS_  16
#define DC_  4
#define DR_  64
#define DK_  (DR_ + 2*DS_)   // 96 rows of x_dbl

// GEMM tiling
#define BM 64
#define BN 32
#define BK 32
#define LDS_PAD 36           // row stride (floats): 144B rows -> 16B aligned, conflict-free wmma reads
#define LDC_PAD 68           // C transpose staging row stride: 272B rows -> 16B aligned
#define TILE_F (BM * LDS_PAD + BN * LDS_PAD)   // 3456 floats per buffer

typedef float v2f __attribute__((ext_vector_type(2)));
typedef float v8f __attribute__((ext_vector_type(8)));

__device__ __forceinline__ float softplus_f(float v) {
    return (v > 20.f) ? v : log1pf(__expf(v));
}
__device__ __forceinline__ float silu_f(float v) {
    return v / (1.f + __expf(-v));
}

// --- gfx1250 async global->LDS 16-byte copy (per-lane), tracked by ASYNCcnt
__device__ __forceinline__ void async_ld16(void* lds, const void* gp) {
    unsigned           l = (unsigned)(uintptr_t)lds;          // low 32b of flat LDS ptr = LDS offset
    unsigned long long g = (unsigned long long)(uintptr_t)gp;
    asm volatile("global_load_async_to_lds_b128 %0, %1, off"
                 :: "v"(l), "v"(g) : "memory");
}
__device__ __forceinline__ void wait_async0() {
#if __has_builtin(__builtin_amdgcn_s_wait_asynccnt)
    __builtin_amdgcn_s_wait_asynccnt(0);
#else
    asm volatile("s_wait_asynccnt 0x0" ::: "memory");
#endif
}

// ----------------------------------------------------------------
// Generic WMMA f32 GEMM:  Ct[b][n][m] = sum_k A[m][k] * Bmat[b][n][k]
// A row-major (lda), Bmat row-major per batch (ldb, strideB),
// output written TRANSPOSED: m contiguous, row stride ldc, batch strideC.
// A rows are CLAMPED to M-1 (stores are guarded), so EXEC stays all-ones.
// EPI: 0 = plain, 1 = softplus(v + bias[m])
// ----------------------------------------------------------------
template <int EPI>
__global__ __launch_bounds__(256) void ssm_wmma_gemm(
    const float* __restrict__ A,
    const float* __restrict__ Bmat,
    float* __restrict__ Ct,
    const float* __restrict__ bias,
    int M, int N, int K,
    int lda, int ldb, long long strideB,
    int ldc, long long strideC)
{
    __shared__ float smem[2 * TILE_F];   // double-buffered A/B tiles; Cs (2176 fl) reuses buf0
    float* Cs = smem;

    const int b    = blockIdx.z;
    const int n0   = blockIdx.x * BN;
    const int m0   = blockIdx.y * BM;
    const int tid  = threadIdx.x;
    const int lane = tid & 31;
    const int wid  = tid >> 5;     // 0..7
    const int wm   = wid & 3;      // sub-tile m: 0..3
    const int wn   = wid >> 2;     // sub-tile n: 0..1
    const int row  = lane & 15;    // M (for A) / N (for B) index within 16
    const int hi   = lane >> 4;    // half-wave select
    const int kh   = hi << 1;      // K offset 0 or 2 per WMMA f32 16x16x4 layout

    const float* Bp = Bmat + (long long)b * strideB;

    // issue one BKxBM + BKxBN tile pair of async global->LDS loads
    auto issue = [&](int buf, int k0) {
        float* As = smem + buf * TILE_F;
        float* Bs = As + BM * LDS_PAD;
        #pragma unroll
        for (int it = 0; it < 2; ++it) {            // A: 64x32 fl = 512 b128, 2/thread
            int c4 = tid + it * 256;
            int r  = c4 >> 3;
            int cc = (c4 & 7) << 2;
            int gr = m0 + r;  gr = (gr < M) ? gr : (M - 1);   // clamp, branch-free
            async_ld16(&As[r * LDS_PAD + cc], A + (long long)gr * lda + k0 + cc);
        }
        int r  = tid >> 3;                           // B: 32x32 fl = 256 b128, 1/thread
        int cc = (tid & 7) << 2;
        async_ld16(&Bs[r * LDS_PAD + cc], Bp + (long long)(n0 + r) * ldb + k0 + cc);
    };

    v8f acc = {0.f, 0.f, 0.f, 0.f, 0.f, 0.f, 0.f, 0.f};

    const int nchunks = K / BK;
    issue(0, 0);
    for (int c = 0; c < nchunks; ++c) {
        const int cur = c & 1;
        wait_async0();                 // drain this wave's async writes into buf[cur]
        __syncthreads();               // make all waves' tile writes visible
        if (c + 1 < nchunks)
            issue(cur ^ 1, (c + 1) * BK);   // prefetch next chunk, overlaps WMMAs

        const float* As = smem + cur * TILE_F;
        const float* Bs = As + BM * LDS_PAD;
        #pragma unroll
        for (int kk = 0; kk < BK; kk += 4) {
            int ka = kk + kh;
            v2f av = *(const v2f*)&As[(wm * 16 + row) * LDS_PAD + ka];
            v2f bv = *(const v2f*)&Bs[(wn * 16 + row) * LDS_PAD + ka];
            acc = __builtin_amdgcn_wmma_f32_16x16x4_f32(
                false, av, false, bv, (short)0, acc, false, false);
        }
        __syncthreads();               // reads done before buf[cur] is refilled next round
    }

    // --- transpose accumulators through LDS for coalesced m-contiguous stores
    #pragma unroll
    for (int j = 0; j < 8; ++j) {
        int ml = wm * 16 + j + (hi ? 8 : 0);
        int nl = wn * 16 + row;
        Cs[nl * LDC_PAD + ml] = acc[j];
    }
    __syncthreads();

    // --- cooperative store: 32 rows x 64 floats = 512 float4, 2 per thread
    #pragma unroll
    for (int it = 0; it < 2; ++it) {
        int c4 = tid + it * 256;               // 0..511
        int rn = c4 >> 4;                      // 0..31
        int m4 = (c4 & 15) << 2;               // 0..60
        int gm = m0 + m4;
        float4 v = *(const float4*)&Cs[rn * LDC_PAD + m4];
        long long off = (long long)b * strideC + (long long)(n0 + rn) * ldc + gm;
        if (gm + 3 < M) {
            if (EPI == 1) {
                float4 bb = *(const float4*)(bias + gm);
                v.x = softplus_f(v.x + bb.x);
                v.y = softplus_f(v.y + bb.y);
                v.z = softplus_f(v.z + bb.z);
                v.w = softplus_f(v.w + bb.w);
            }
            *(float4*)(Ct + off) = v;
        } else {
            float vv[4] = {v.x, v.y, v.z, v.w};
            #pragma unroll
            for (int i = 0; i < 4; ++i) {
                if (gm + i < M) {
                    float o = vv[i];
                    if (EPI == 1) o = softplus_f(o + bias[gm + i]);
                    Ct[off + i] = o;
                }
            }
        }
    }
}

// ----------------------------------------------------------------
// Depthwise causal conv (width 4) + bias + SiLU.
// x_t, u: [B][L][DI] (e contiguous)
// ----------------------------------------------------------------
__global__ __launch_bounds__(256) void ssm_conv_silu(
    const float* __restrict__ x_t,
    const float* __restrict__ conv_w,   // [DI][DC]
    const float* __restrict__ conv_b,   // [DI]
    float* __restrict__ u)
{
    long long idx = (long long)blockIdx.x * 256 + threadIdx.x;  // e fastest
    int e = (int)(idx % DI_);
    long long rest = idx / DI_;
    int l = (int)(rest % L_);
    int b = (int)(rest / L_);

    float acc = conv_b[e];
    const float* xb = x_t + (long long)b * L_ * DI_;
    #pragma unroll
    for (int j = 0; j < DC_; ++j) {
        int ll = l - (DC_ - 1) + j;
        if (ll >= 0)
            acc += conv_w[e * DC_ + j] * xb[(long long)ll * DI_ + e];
    }
    u[idx] = silu_f(acc);
}

// ----------------------------------------------------------------
// Selective scan. Wave = 2 channels x 16 states (n).
// lanes 0-15: channel 2p, n=lane; lanes 16-31: channel 2p+1, n=lane-16.
// delta_t, u, out: [B][L][DI]; xdbl_t: [B][L][96]; z: [B][DI][L]
// ----------------------------------------------------------------
__global__ __launch_bounds__(256) void ssm_scan(
    const float* __restrict__ delta_t,
    const float* __restrict__ u,
    const float* __restrict__ xdbl_t,
    const float* __restrict__ z,
    const float* __restrict__ A_log,    // [DI][DS]
    const float* __restrict__ Dvec,     // [DI]
    float* __restrict__ out)
{
    int gwave = (int)((blockIdx.x * blockDim.x + threadIdx.x) >> 5);
    int lane  = threadIdx.x & 31;
    int hi    = lane >> 4;
    int n     = lane & 15;

    int pair = gwave % (DI_ / 2);
    int b    = gwave / (DI_ / 2);
    int e    = pair * 2 + hi;

    float a_coef = -__expf(A_log[e * DS_ + n]);
    float d_e    = Dvec[e];

    const float* dp = delta_t + (long long)b * L_ * DI_ + e;
    const float* up = u       + (long long)b * L_ * DI_ + e;
    const float* xb = xdbl_t  + (long long)b * L_ * DK_;
    const float* zp = z + ((long long)b * DI_ + e) * L_;
    float* op = out + (long long)b * L_ * DI_ + e;

    float s = 0.f;
    for (int t = 0; t < L_; ++t) {
        float dlt = dp[(long long)t * DI_];
        float ut  = up[(long long)t * DI_];
        float Btn = xb[t * DK_ + DR_ + n];
        float Ctn = xb[t * DK_ + DR_ + DS_ + n];

        s = s * __expf(dlt * a_coef) + (dlt * ut) * Btn;

        float part = s * Ctn;
        #pragma unroll
        for (int off = 1; off < 16; off <<= 1)
            part += __shfl_xor(part, off, 16);

        if (n == 0) {
            float y  = part + d_e * ut;
            float zv = zp[t];
            op[(long long)t * DI_] = y * silu_f(zv);
        }
    }
}

// ---------------------------------------------------------------- launch
extern "C" void kernel_launch(void* const* d_in, const int* in_sizes, int n_in,
                              void* d_out, int out_size, void* d_ws, size_t ws_size,
                              hipStream_t stream) {
    const float* hs        = (const float*)d_in[0];  // [B][L][DM]
    const float* z         = (const float*)d_in[1];  // [B][DI][L]
    const float* in_proj_w = (const float*)d_in[2];  // [DI][DM]
    const float* conv_w    = (const float*)d_in[3];  // [DI][DC]
    const float* conv_b    = (const float*)d_in[4];  // [DI]
    const float* x_proj_w  = (const float*)d_in[5];  // [96][DI]
    const float* dt_proj_w = (const float*)d_in[6];  // [DI][DR]
    const float* dt_proj_b = (const float*)d_in[7];  // [DI]
    const float* A_log     = (const float*)d_in[8];  // [DI][DS]
    const float* Dv        = (const float*)d_in[9];  // [DI]
    float* out = (float*)d_out;

    float* ws      = (float*)d_ws;
    float* x_t     = ws;                                       // [B][L][DI]
    float* u_le    = x_t    + (size_t)B_ * L_ * DI_;           // [B][L][DI]
    float* xdbl_t  = u_le   + (size_t)B_ * L_ * DI_;           // [B][L][96]
    float* delta_t = xdbl_t + (size_t)B_ * L_ * DK_;           // [B][L][DI]

    dim3 blk(256);

    // K1: x_t[b][l][e] = sum_d in_proj_w[e][d] * hs[b][l][d]
    ssm_wmma_gemm<0><<<dim3(L_ / BN, DI_ / BM, B_), blk, 0, stream>>>(
        in_proj_w, hs, x_t, nullptr,
        DI_, L_, DM_, DM_, DM_, (long long)L_ * DM_, DI_, (long long)L_ * DI_);

    // K2: depthwise conv + SiLU
    ssm_conv_silu<<<dim3((B_ * L_ * DI_) / 256), blk, 0, stream>>>(
        x_t, conv_w, conv_b, u_le);

    // K3: xdbl_t[b][l][k] = sum_e x_proj_w[k][e] * u_le[b][l][e]   (M=96)
    ssm_wmma_gemm<0><<<dim3(L_ / BN, (DK_ + BM - 1) / BM, B_), blk, 0, stream>>>(
        x_proj_w, u_le, xdbl_t, nullptr,
        DK_, L_, DI_, DI_, DI_, (long long)L_ * DI_, DK_, (long long)L_ * DK_);

    // K4: delta_t[b][l][e] = softplus(sum_r dt_proj_w[e][r]*xdbl_t[b][l][r] + dt_proj_b[e])
    ssm_wmma_gemm<1><<<dim3(L_ / BN, DI_ / BM, B_), blk, 0, stream>>>(
        dt_proj_w, xdbl_t, delta_t, dt_proj_b,
        DI_, L_, DR_, DR_, DK_, (long long)L_ * DK_, DI_, (long long)L_ * DI_);

    // K5: selective scan + D*u + silu(z) gating, writes [B][L][DI]
    ssm_scan<<<dim3((B_ * (DI_ / 2) * 32) / 256), blk, 0, stream>>>(
        delta_t, u_le, xdbl_t, z, A_log, Dv, out);
}